// ReinforcementModel_15212774162972
// MI455X (gfx1250) — compile-verified
//
#include <hip/hip_runtime.h>
#include <math.h>
#include <stdint.h>

// One wave32 handles one batch row:
//  - async-DMA the 128-float "next" row (512 B) into LDS (ASYNCcnt path)
//  - concurrently gather action / scalar logit / value / mask
//  - s_wait_asynccnt 0, read float4/lane from LDS, 5-step shfl_xor max tree
//  - lane 0 writes tanh(scalar) and the masked target.

#define LANES_PER_WAVE 32
#define WAVES_PER_BLOCK 8
#define ACT_DIM 128   // A: one row == 32 lanes * float4

__global__ __launch_bounds__(WAVES_PER_BLOCK * LANES_PER_WAVE)
void reinforce_gather_max_kernel(const float* __restrict__ logits,
                                 const float* __restrict__ values,
                                 const int*   __restrict__ actions,
                                 const int*   __restrict__ sample_ind,
                                 const int*   __restrict__ values_mask,
                                 float* __restrict__ out,
                                 int B, int L)
{
    __shared__ float row_lds[WAVES_PER_BLOCK * ACT_DIM];

    const int tid  = threadIdx.x;
    const int wave = tid / LANES_PER_WAVE;
    const int lane = tid % LANES_PER_WAVE;
    const int b    = blockIdx.x * WAVES_PER_BLOCK + wave;
    if (b >= B) return;

    const int    s       = sample_ind[b];
    const size_t rowbase = ((size_t)b * (size_t)L + (size_t)s) * (size_t)ACT_DIM;

    // ---- kick off async DMA of next row (s+1) into this wave's LDS slice ----
    // Each lane moves 16 bytes: 32 lanes * 16 B = 512 B = one 128-float row.
    // ldsoff is derived from the real pointer (ptrtoint): this both escapes
    // row_lds (so the "memory" clobber legally covers the async write) and
    // matches the ISA rule that a flat LDS pointer's low 32 bits are the
    // workgroup-relative LDS byte address the async VDST operand expects.
    const float*   gsrc   = logits + rowbase + ACT_DIM + (size_t)(lane * 4);
    float*         lptr   = &row_lds[wave * ACT_DIM + lane * 4];
    const unsigned ldsoff = (unsigned)(uintptr_t)lptr;
    asm volatile("global_load_async_to_lds_b128 %0, %1, off"
                 :: "v"(ldsoff), "v"(gsrc)
                 : "memory");

    // ---- overlap: scalar gathers (broadcast loads, same addr per wave) ----
    const int   act = actions[(size_t)b * (size_t)L + (size_t)s];
    const float x   = logits[rowbase + (size_t)act];   // logits[b, s, act]
    const float vb  = values[b];
    const int   mb  = values_mask[b];

    // ---- wait for the async row DMA, then reduce from LDS ----
    asm volatile("s_wait_asynccnt 0" ::: "memory");
    const float4 v4 = *reinterpret_cast<const float4*>(lptr);

    float m = fmaxf(fmaxf(v4.x, v4.y), fmaxf(v4.z, v4.w));
    #pragma unroll
    for (int off = LANES_PER_WAVE / 2; off > 0; off >>= 1)
        m = fmaxf(m, __shfl_xor(m, off, LANES_PER_WAVE));

    if (lane == 0) {
        out[b]     = tanhf(x);              // est_values
        out[B + b] = mb ? vb : m;           // targets
    }
}

extern "C" void kernel_launch(void* const* d_in, const int* in_sizes, int n_in,
                              void* d_out, int out_size, void* d_ws, size_t ws_size,
                              hipStream_t stream) {
    const float* logits      = (const float*)d_in[0];
    const float* values      = (const float*)d_in[1];
    const int*   actions     = (const int*)d_in[2];
    const int*   sample_ind  = (const int*)d_in[3];
    const int*   values_mask = (const int*)d_in[4];
    float*       out         = (float*)d_out;

    const int B = in_sizes[1];              // values: (B,)
    const int L = in_sizes[2] / B;          // actions: (B, L)
    // in_sizes[0] / (B*L) == 128 == ACT_DIM (layout assumption: 32 lanes x float4)

    const int blocks = (B + WAVES_PER_BLOCK - 1) / WAVES_PER_BLOCK;
    reinforce_gather_max_kernel<<<blocks, WAVES_PER_BLOCK * LANES_PER_WAVE, 0, stream>>>(
        logits, values, actions, sample_ind, values_mask, out, B, L);
}